// RGBICPPoseEstimator_40819369181723
// MI455X (gfx1250) — compile-verified
//
#include <hip/hip_runtime.h>
#include <math.h>

#define HH 768
#define WW 1024
#define NPIX (HH*WW)
#define FXC 1000.0f
#define FYC 1000.0f
#define CXC 512.0f
#define CYC 384.0f
#define DIST_THR2 ((200.0f/15.0f)*(200.0f/15.0f))
#define NORMAL_COS 0.93969262078590838f
#define SQRT10 3.16227766016837933f
#define NBLOCKS 512
#define NITER 10

typedef float v2f __attribute__((ext_vector_type(2)));
typedef float v8f __attribute__((ext_vector_type(8)));

struct F3 { float x, y, z; };
__device__ __forceinline__ F3 mkf3(float a, float b, float c) { F3 r; r.x=a; r.y=b; r.z=c; return r; }
__device__ __forceinline__ F3 f3sub(F3 a, F3 b) { return mkf3(a.x-b.x, a.y-b.y, a.z-b.z); }
__device__ __forceinline__ F3 f3crs(F3 a, F3 b) {
    return mkf3(a.y*b.z - a.z*b.y, a.z*b.x - a.x*b.z, a.x*b.y - a.y*b.x);
}
__device__ __forceinline__ float f3dot(F3 a, F3 b) { return a.x*b.x + a.y*b.y + a.z*b.z; }

__device__ __forceinline__ F3 bproj(const float* __restrict__ dep, int vv, int uu) {
    float d = dep[vv*WW + uu];
    return mkf3((uu - CXC) * (1.0f/FXC) * d, (vv - CYC) * (1.0f/FYC) * d, d);
}

// R = exp([w]x), row major, matches reference so3_exp (incl. 1e-12 eps)
__device__ __forceinline__ void so3R(const float* w, float* R) {
    float th = sqrtf(w[0]*w[0] + w[1]*w[1] + w[2]*w[2] + 1e-12f);
    float inv = 1.0f / th;
    float kx = w[0]*inv, ky = w[1]*inv, kz = w[2]*inv;
    float s = sinf(th), c = 1.0f - cosf(th);
    R[0] = 1.0f + c*(kx*kx - 1.0f); R[1] = -s*kz + c*kx*ky;        R[2] =  s*ky + c*kx*kz;
    R[3] =  s*kz + c*ky*kx;         R[4] = 1.0f + c*(ky*ky - 1.0f); R[5] = -s*kx + c*ky*kz;
    R[6] = -s*ky + c*kz*kx;         R[7] =  s*kx + c*kz*ky;        R[8] = 1.0f + c*(kz*kz - 1.0f);
}

// one-time central-difference images (roll semantics -> wrap); iteration-invariant
__global__ void __launch_bounds__(256) grad_prep(const float* __restrict__ g,
                                                 float* __restrict__ gx,
                                                 float* __restrict__ gy) {
    int i = blockIdx.x * 256 + threadIdx.x;
    if (i >= NPIX) return;
    int tu = i & (WW - 1);
    int tv = i >> 10;
    int up1 = (tu + 1) & (WW - 1);
    int um1 = (tu - 1) & (WW - 1);
    int vp1 = (tv + 1 == HH) ? 0 : tv + 1;
    int vm1 = (tv == 0) ? HH - 1 : tv - 1;
    gx[i] = 0.5f * (g[tv*WW + up1] - g[tv*WW + um1]);
    gy[i] = 0.5f * (g[vp1*WW + tu] - g[vm1*WW + tu]);
}

// state layout: [0:6]=x  [6:15]=R  [15:21]=best_x  [21]=best_cost
__global__ void icp_init(const float* __restrict__ init_x, float* __restrict__ state) {
    if (threadIdx.x == 0 && blockIdx.x == 0) {
        float x[6];
        #pragma unroll
        for (int i = 0; i < 6; ++i) { x[i] = init_x[i]; state[i] = x[i]; state[15+i] = x[i]; }
        state[21] = __int_as_float(0x7f800000); // +inf
        so3R(x, state + 6);
    }
}

__global__ void __launch_bounds__(256) icp_pass(
    const float* __restrict__ ref_depth, const float* __restrict__ ref_gray,
    const float* __restrict__ tgt_gray,  const float* __restrict__ tgt_pts,
    const float* __restrict__ tgt_nrm,   const unsigned char* __restrict__ rmask,
    const unsigned char* __restrict__ tmask, const float* __restrict__ gximg,
    const float* __restrict__ gyimg,     const float* __restrict__ state,
    float* __restrict__ partial)
{
    __shared__ float stage[8][32][17];   // per-wave 32 points x 16 comps (pad 17: bank-coprime)
    __shared__ float gram[8][16][16];    // per-wave 16x16 Gram dump

    const int wave = threadIdx.x >> 5;
    const int lane = threadIdx.x & 31;
    const int col  = lane & 15;
    const int totalWaves = gridDim.x * 8;
    const int gw = blockIdx.x * 8 + wave;

    // uniform pose
    const float R0 = state[6],  R1 = state[7],  R2 = state[8];
    const float R3 = state[9],  R4 = state[10], R5 = state[11];
    const float R6 = state[12], R7 = state[13], R8 = state[14];
    const float tx = state[3],  ty = state[4],  tz = state[5];

    v8f acc;
    #pragma unroll
    for (int i = 0; i < 8; ++i) acc[i] = 0.0f;

    for (int base = gw * 32; base < NPIX; base += totalWaves * 32) {
        int idx = base + lane;
        bool live = idx < NPIX;
        int id = live ? idx : 0;
        int pu = id & (WW - 1);
        int pv = id >> 10;

        // --- ref point + normal (recomputed, L2-resident) ---
        F3 P0 = bproj(ref_depth, pv, pu);
        F3 Pr = bproj(ref_depth, pv, (pu + 1) & (WW - 1));
        F3 Pd = bproj(ref_depth, (pv + 1 == HH) ? 0 : pv + 1, pu);
        F3 nr = f3crs(f3sub(Pr, P0), f3sub(Pd, P0));
        float inl = rsqrtf(f3dot(nr, nr) + 1e-12f);
        nr = mkf3(nr.x*inl, nr.y*inl, nr.z*inl);

        // --- transform + project ---
        F3 p = mkf3(R0*P0.x + R1*P0.y + R2*P0.z + tx,
                    R3*P0.x + R4*P0.y + R5*P0.z + ty,
                    R6*P0.x + R7*P0.y + R8*P0.z + tz);
        float z = p.z;
        float u = FXC * p.x / z + CXC;
        float v = FYC * p.y / z + CYC;
        bool inb = (u >= 0.0f) & (u <= (float)(WW-1)) & (v >= 0.0f) & (v <= (float)(HH-1)) & (z > 1e-6f);
        float uc = inb ? u : 0.0f;
        float vc = inb ? v : 0.0f;

        // --- nearest-neighbor association (round half-to-even == rintf) ---
        int ui = min(max((int)rintf(uc), 0), WW - 1);
        int vi = min(max((int)rintf(vc), 0), HH - 1);
        int flat = vi * WW + ui;
        F3 q  = mkf3(tgt_pts[3*flat+0], tgt_pts[3*flat+1], tgt_pts[3*flat+2]);
        F3 nq = mkf3(tgt_nrm[3*flat+0], tgt_nrm[3*flat+1], tgt_nrm[3*flat+2]);
        F3 diff = f3sub(p, q);
        F3 rn = mkf3(R0*nr.x + R1*nr.y + R2*nr.z,
                     R3*nr.x + R4*nr.y + R5*nr.z,
                     R6*nr.x + R7*nr.y + R8*nr.z);

        bool rm = rmask[id] != 0;
        bool tm = tmask[flat] != 0;
        bool common = inb & rm & tm;
        bool vicp = common & (f3dot(diff, diff) < DIST_THR2) & (f3dot(rn, nq) > NORMAL_COS);
        float w_icp = (vicp & live) ? 1.0f : 0.0f;
        float w_rgb = (common & live) ? 1.0f : 0.0f;

        // --- bilinear taps of gray / precomputed gx,gy ---
        float u0f = floorf(uc), v0f = floorf(vc);
        float du = uc - u0f, dv = vc - v0f;
        int u0 = min(max((int)u0f, 0), WW - 1);
        int u1 = min(u0 + 1, WW - 1);
        int v0 = min(max((int)v0f, 0), HH - 1);
        int v1 = min(v0 + 1, HH - 1);
        int t00 = v0*WW + u0, t01 = v0*WW + u1, t10 = v1*WW + u0, t11 = v1*WW + u1;
        float w00 = (1.0f-du)*(1.0f-dv), w01 = du*(1.0f-dv);
        float w10 = (1.0f-du)*dv,        w11 = du*dv;
        float Ig  = tgt_gray[t00]*w00 + tgt_gray[t01]*w01 + tgt_gray[t10]*w10 + tgt_gray[t11]*w11;
        float gxs = gximg[t00]*w00 + gximg[t01]*w01 + gximg[t10]*w10 + gximg[t11]*w11;
        float gys = gyimg[t00]*w00 + gyimg[t01]*w01 + gyimg[t10]*w10 + gyimg[t11]*w11;

        float zg = (z > 1e-6f) ? z : 1.0f;  // guard: masked lanes must stay finite
        F3 g = mkf3(gxs * FXC / zg, gys * FYC / zg,
                    -(gxs * FXC * p.x + gys * FYC * p.y) / (zg * zg));

        // --- stage 16-component V into LDS (Gram vector) ---
        float swi = SQRT10 * w_icp;
        F3 cpn = f3crs(p, nq);
        F3 cpg = f3crs(p, g);
        float* sp = &stage[wave][lane][0];
        sp[0]  = swi * cpn.x;  sp[1]  = swi * cpn.y;  sp[2]  = swi * cpn.z;
        sp[3]  = swi * nq.x;   sp[4]  = swi * nq.y;   sp[5]  = swi * nq.z;
        sp[6]  = swi * f3dot(nq, diff);
        sp[7]  = w_rgb * cpg.x; sp[8] = w_rgb * cpg.y; sp[9] = w_rgb * cpg.z;
        sp[10] = w_rgb * g.x;  sp[11] = w_rgb * g.y;  sp[12] = w_rgb * g.z;
        sp[13] = (Ig - ref_gray[id]) * w_rgb;
        sp[14] = w_icp;
        sp[15] = 0.0f;

        // make same-wave LDS stores visible before transposed reads
        asm volatile("s_wait_dscnt 0" ::: "memory");

        // --- load all 8 operand pairs first (distinct regs -> overlapped DS waits),
        //     then chain 8x V_WMMA_F32_16X16X4_F32 Gram accumulates (4 points each) ---
        v2f av[8];
        #pragma unroll
        for (int gq = 0; gq < 8; ++gq) {
            int row = 4*gq + ((lane < 16) ? 0 : 2);
            av[gq].x = stage[wave][row    ][col];
            av[gq].y = stage[wave][row + 1][col];
        }
        #pragma unroll
        for (int gq = 0; gq < 8; ++gq) {
            acc = __builtin_amdgcn_wmma_f32_16x16x4_f32(
                false, av[gq], false, av[gq], (short)0, acc, false, false);
        }
        // no trailing drain needed: same-wave LDS ops are processed in order,
        // so next pass's ds_stores cannot bypass this pass's ds_loads.
    }

    // dump distributed 16x16 accumulator: vgpr g @ lane l -> (M = g + (l<16?0:8), N = l%16)
    int mbase = (lane < 16) ? 0 : 8;
    #pragma unroll
    for (int gq = 0; gq < 8; ++gq) gram[wave][mbase + gq][col] = acc[gq];
    __syncthreads();

    int e = threadIdx.x;           // 256 Gram entries, one per thread
    float s = 0.0f;
    #pragma unroll
    for (int wv = 0; wv < 8; ++wv) s += gram[wv][e >> 4][e & 15];
    partial[blockIdx.x * 256 + e] = s;
}

__global__ void __launch_bounds__(256) icp_solve(const float* __restrict__ partial,
                                                 float* __restrict__ state,
                                                 float* __restrict__ out)
{
    __shared__ float G[256];
    int t = threadIdx.x;
    float s = 0.0f;
    for (int b = 0; b < NBLOCKS; ++b) s += partial[b * 256 + t];
    G[t] = s;
    __syncthreads();

    if (t == 0) {
        // A = G[0:6,0:6] + G[7:13,7:13]; b = G[:,6] + G[:,13]; ICP weight folded via sqrt(10)
        float A[6][7];
        #pragma unroll
        for (int i = 0; i < 6; ++i) {
            #pragma unroll
            for (int j = 0; j < 6; ++j)
                A[i][j] = G[i*16 + j] + G[(7+i)*16 + (7+j)];
            A[i][i] += 1e-6f;
            A[i][6] = G[i*16 + 6] + G[(7+i)*16 + 13];
        }
        float nicp = fmaxf(G[14*16 + 14], 1.0f);
        float cost = (G[6*16 + 6] + G[13*16 + 13]) / nicp;

        if (cost < state[21]) {          // best tracks the *pre-update* x
            state[21] = cost;
            #pragma unroll
            for (int i = 0; i < 6; ++i) state[15+i] = state[i];
        }

        // Gauss-Jordan on SPD 6x6 (+ridge)
        #pragma unroll
        for (int k = 0; k < 6; ++k) {
            float inv = 1.0f / A[k][k];
            #pragma unroll
            for (int j = 0; j < 7; ++j) A[k][j] *= inv;
            #pragma unroll
            for (int r = 0; r < 6; ++r) {
                if (r == k) continue;
                float f = A[r][k];
                #pragma unroll
                for (int j = 0; j < 7; ++j) A[r][j] -= f * A[k][j];
            }
        }
        float xn[6];
        #pragma unroll
        for (int i = 0; i < 6; ++i) { xn[i] = state[i] - A[i][6]; state[i] = xn[i]; }
        so3R(xn, state + 6);             // pose for next pass

        #pragma unroll
        for (int i = 0; i < 6; ++i) out[i] = state[15+i];
    }
}

extern "C" void kernel_launch(void* const* d_in, const int* in_sizes, int n_in,
                              void* d_out, int out_size, void* d_ws, size_t ws_size,
                              hipStream_t stream) {
    (void)in_sizes; (void)n_in; (void)out_size; (void)ws_size;
    const float* ref_depth = (const float*)d_in[0];
    const float* ref_gray  = (const float*)d_in[1];
    const float* tgt_gray  = (const float*)d_in[2];
    const float* tgt_pts   = (const float*)d_in[3];
    const float* tgt_nrm   = (const float*)d_in[4];
    const unsigned char* rmask = (const unsigned char*)d_in[5];
    const unsigned char* tmask = (const unsigned char*)d_in[6];
    const float* init_x    = (const float*)d_in[7];
    float* out = (float*)d_out;

    float* gx      = (float*)d_ws;                 // NPIX floats
    float* gy      = gx + NPIX;                    // NPIX floats
    float* partial = gy + NPIX;                    // NBLOCKS*256 floats
    float* state   = partial + NBLOCKS * 256;      // 22 floats

    grad_prep<<<(NPIX + 255) / 256, 256, 0, stream>>>(tgt_gray, gx, gy);
    icp_init<<<1, 1, 0, stream>>>(init_x, state);
    for (int it = 0; it < NITER; ++it) {
        icp_pass<<<NBLOCKS, 256, 0, stream>>>(ref_depth, ref_gray, tgt_gray,
                                              tgt_pts, tgt_nrm, rmask, tmask,
                                              gx, gy, state, partial);
        icp_solve<<<1, 256, 0, stream>>>(partial, state, out);
    }
}